// DenseDilatedKnnGraph_53661321396520
// MI455X (gfx1250) — compile-verified
//
#include <hip/hip_runtime.h>
#include <stdint.h>

// ---------------------------------------------------------------------------
// DenseDilatedKnnGraph on MI455X (gfx1250, wave32, WMMA + TDM).
//
// 68.7 GFLOP f32 Gram matrix vs ~50MB HBM -> matrix-pipe bound. NxN distances
// never materialized; top-18 fused into WMMA epilogue with (distBits,idx)
// ordering identical to jax.lax.top_k in f32 (index output must be exact, so
// bf16/fp8 WMMA is rejected: it would flip near-tie neighbors).
//
// B tile staged by the Tensor Data Mover (tensor_load_to_lds, TENSORcnt,
// double-buffered, hardware LDS padding = 260-dword rows), one barrier per
// chunk, two independent WMMA accumulator chains.
// Toolchain probe result: 6-arg tensor_load_to_lds (clang-23 / therock form).
// ---------------------------------------------------------------------------

typedef float        v2f __attribute__((ext_vector_type(2)));
typedef float        v8f __attribute__((ext_vector_type(8)));
typedef unsigned int v4u __attribute__((ext_vector_type(4)));
typedef int          v4i __attribute__((ext_vector_type(4)));
typedef int          v8i __attribute__((ext_vector_type(8)));

#define B_BATCH 2
#define C_DIM   256
#define N_PTS   8192
#define K_OUT   9
#define K_TOTAL 18

#define WAVES_PER_BLOCK 8
#define ROWS_PER_WAVE   16
#define ROWS_PER_BLOCK  (WAVES_PER_BLOCK * ROWS_PER_WAVE)   // 128
#define THREADS         256                                  // 8 x wave32
#define KFRAGS          (C_DIM / 4)                          // 64 WMMA k-steps
#define BP_STRIDE       260   // 256 data + 4 pad dwords (TDM pad_interval=7,
                              // pad_amount=3) -> bank-clean ds_load_b64 frags

// ---------------------------------------------------------------------------
// Phase 1: L2-normalize along C, transpose (B,C,N) -> (B,N,C), store
// sq[b][n] = sum(xn^2) of the *rounded* normalized values (as the ref does).
// ---------------------------------------------------------------------------
__global__ __launch_bounds__(THREADS)
void knn_normalize_kernel(const float* __restrict__ x,
                          float* __restrict__ xt,
                          float* __restrict__ sq) {
    const int g = blockIdx.x * THREADS + threadIdx.x;   // b*N + n
    const int n = g & (N_PTS - 1);
    const int b = g >> 13;
    const float* xp = x + (size_t)b * C_DIM * N_PTS + n;

    float ss = 0.0f;
    for (int c = 0; c < C_DIM; ++c) {
        float v = xp[(size_t)c * N_PTS];
        ss += v * v;
    }
    const float inv = 1.0f / fmaxf(sqrtf(ss), 1e-12f);

    float* dst = xt + (size_t)g * C_DIM;
    float ss2 = 0.0f;
    for (int c = 0; c < C_DIM; ++c) {
        float v = xp[(size_t)c * N_PTS] * inv;
        dst[c] = v;
        ss2 += v * v;
    }
    sq[g] = ss2;
}

// ---------------------------------------------------------------------------
// Phase 2: TDM-staged, double-buffered f32 WMMA Gram + fused top-18.
// ---------------------------------------------------------------------------
__global__ __launch_bounds__(THREADS)
void knn_gemm_topk_kernel(const float* __restrict__ xt,
                          const float* __restrict__ sq,
                          int* __restrict__ out) {
    const int tid  = threadIdx.x;
    const int wave = tid >> 5;
    const int lane = tid & 31;
    const int p    = lane & 15;   // point slot within 16
    const int h    = lane >> 4;   // half-wave -> k-pair select

    const int blocksPerBatch = N_PTS / ROWS_PER_BLOCK;          // 64
    const int b       = blockIdx.x / blocksPerBatch;
    const int rowBase = (blockIdx.x % blocksPerBatch) * ROWS_PER_BLOCK
                        + wave * ROWS_PER_WAVE;

    const float* xtb = xt + (size_t)b * N_PTS * C_DIM;
    const float* sqb = sq + (size_t)b * N_PTS;

    __shared__ float Bp[2][16 * BP_STRIDE];                 // TDM-filled B tiles
    __shared__ float keybuf[WAVES_PER_BLOCK][16 * 17];      // distances [row][col]
    __shared__ uint64_t topk[WAVES_PER_BLOCK][16 * K_TOTAL];

    // ---- TDM descriptor issue (wave 0 only; EXEC-independent DMA) ----------
    // D# group0: count=1, lds_addr, 57b global_addr, type=2.
    // D# group1: data_size=4B, pad_enable, pad_interval=7 (256 dw),
    //            pad_amount=3 (4 dw), tensor=(256 x 8192) stride 256,
    //            tile=(256 x 16).
    auto issue_tdm = [&](int colBase, int buf) {
        const uint64_t ga  = (uint64_t)(uintptr_t)(xtb + (size_t)colBase * C_DIM);
        const uint32_t lds = (uint32_t)(uintptr_t)&Bp[buf][0]; // LDS aperture low 32b
        v4u g0 = { 1u,                                   // count=1 (valid D#)
                   lds,
                   (uint32_t)ga,
                   (uint32_t)((ga >> 32) & 0x01FFFFFFu) | 0x80000000u };
        v8i g1 = { (int)((2u << 16) | (1u << 20) | (7u << 22) | (3u << 25)),
                   (int)(256u << 16),                    // tensor_dim0 = 256
                   (int)(8192u << 16),                   // tensor_dim1 = 8192
                   (int)(256u << 16),                    // tile_dim0 = 256
                   16,                                   // tile_dim1 = 16
                   256,                                  // tensor_dim0_stride
                   0, 0 };
        v4i g2 = { 0, 0, 0, 0 };
        v4i g3 = { 0, 0, 0, 0 };
        v8i g4 = { 0, 0, 0, 0, 0, 0, 0, 0 };
        __builtin_amdgcn_tensor_load_to_lds(g0, g1, g2, g3, g4, 0);
    };

    // ---- A fragments resident in VGPRs: af[f] = xt[row][4f+2h .. +1] -------
    const int rowp = rowBase + p;
    v2f af[KFRAGS];
    #pragma unroll
    for (int f = 0; f < KFRAGS; ++f)
        af[f] = *(const v2f*)(xtb + (size_t)rowp * C_DIM + 4 * f + 2 * h);

    float sqrow[8];
    #pragma unroll
    for (int r = 0; r < 8; ++r)
        sqrow[r] = sqb[rowBase + r + 8 * h];

    // ---- per-row selection state (lane l<16 owns local row l) --------------
    uint64_t curMax = ~0ull;
    int      maxPos = 0;
    if (lane < 16) {
        #pragma unroll
        for (int j = 0; j < K_TOTAL; ++j)
            topk[wave][lane * K_TOTAL + j] = ~0ull;
    }

    if (wave == 0) issue_tdm(0, 0);    // prime buffer 0

    for (int chunk = 0; chunk < N_PTS / 16; ++chunk) {
        const int colBase = chunk * 16;
        const int cur     = chunk & 1;

        if (wave == 0) __builtin_amdgcn_s_wait_tensorcnt(0);  // tile landed
        __syncthreads();   // publish Bp[cur]; all waves done with Bp[cur^1]

        if (wave == 0 && chunk + 1 < N_PTS / 16)
            issue_tdm(colBase + 16, cur ^ 1);                 // overlap next DMA

        // ---- 16x16 Gram tile: two independent WMMA accumulation chains -----
        v8f acc0 = {0.f,0.f,0.f,0.f,0.f,0.f,0.f,0.f};
        v8f acc1 = {0.f,0.f,0.f,0.f,0.f,0.f,0.f,0.f};
        const float* bptr = &Bp[cur][p * BP_STRIDE + 2 * h];
        #pragma unroll
        for (int f = 0; f < KFRAGS; f += 2) {
            v2f bf0 = *(const v2f*)(bptr + 4 * f);
            v2f bf1 = *(const v2f*)(bptr + 4 * f + 4);
            acc0 = __builtin_amdgcn_wmma_f32_16x16x4_f32(
                false, af[f],     false, bf0, (short)0, acc0, false, false);
            acc1 = __builtin_amdgcn_wmma_f32_16x16x4_f32(
                false, af[f + 1], false, bf1, (short)0, acc1, false, false);
        }

        // ---- epilogue: d = sqrt(max(sq_m + sq_n - 2*dot, 0)) ---------------
        {
            const float sc = sqb[colBase + p];     // sq is L2-resident (32KB)
            float* kb = keybuf[wave];
            #pragma unroll
            for (int r = 0; r < 8; ++r) {
                float d2 = sqrow[r] + sc - 2.0f * (acc0[r] + acc1[r]);
                kb[(r + 8 * h) * 17 + p] = sqrtf(fmaxf(d2, 0.0f));
            }
        }

        // ---- per-row top-18 maintenance (replace-max, rare rescans) --------
        if (lane < 16) {
            uint64_t* tk = &topk[wave][lane * K_TOTAL];
            const float* krow = &keybuf[wave][lane * 17];
            #pragma unroll 1
            for (int c = 0; c < 16; ++c) {
                uint64_t key = ((uint64_t)__float_as_uint(krow[c]) << 32)
                             | (uint32_t)(colBase + c);
                if (key < curMax) {
                    tk[maxPos] = key;
                    uint64_t m = tk[0]; int mp = 0;
                    #pragma unroll
                    for (int j = 1; j < K_TOTAL; ++j) {
                        uint64_t v = tk[j];
                        if (v > m) { m = v; mp = j; }
                    }
                    curMax = m; maxPos = mp;
                }
            }
        }
    }

    // ---- final: sort 18 keys ascending, emit dilated slice ::2 (K=9) -------
    if (lane < 16) {
        uint64_t* tk = &topk[wave][lane * K_TOTAL];
        for (int i = 1; i < K_TOTAL; ++i) {
            uint64_t v = tk[i];
            int j = i - 1;
            while (j >= 0 && tk[j] > v) { tk[j + 1] = tk[j]; --j; }
            tk[j + 1] = v;
        }
        const int n = rowBase + lane;
        int* out0 = out + ((size_t)(0 * B_BATCH + b) * N_PTS + n) * K_OUT;
        int* out1 = out + ((size_t)(1 * B_BATCH + b) * N_PTS + n) * K_OUT;
        #pragma unroll
        for (int j = 0; j < K_OUT; ++j) {
            out0[j] = (int)(uint32_t)tk[2 * j];   // low 32 bits = neighbor idx
            out1[j] = n;
        }
    }
}

// ---------------------------------------------------------------------------
extern "C" void kernel_launch(void* const* d_in, const int* in_sizes, int n_in,
                              void* d_out, int out_size, void* d_ws, size_t ws_size,
                              hipStream_t stream) {
    const float* x = (const float*)d_in[0];     // (2, 256, 8192, 1) f32
    float* xt = (float*)d_ws;                   // 16 MB (B,N,C)
    float* sq = (float*)((char*)d_ws + (size_t)B_BATCH * N_PTS * C_DIM * sizeof(float));
    int*   out = (int*)d_out;                   // (2, 2, 8192, 9) int32

    knn_normalize_kernel<<<(B_BATCH * N_PTS) / THREADS, THREADS, 0, stream>>>(x, xt, sq);
    knn_gemm_topk_kernel<<<B_BATCH * (N_PTS / ROWS_PER_BLOCK), THREADS, 0, stream>>>(xt, sq, out);
}